// RelationConv_32667521253700
// MI455X (gfx1250) — compile-verified
//
#include <hip/hip_runtime.h>
#include <hip/hip_bf16.h>
#include <math.h>

typedef __attribute__((ext_vector_type(2))) float v2f;
typedef __attribute__((ext_vector_type(8))) float v8f;
typedef __attribute__((ext_vector_type(4))) unsigned int u32x4;
typedef __attribute__((ext_vector_type(8))) int i32x8;
typedef __attribute__((ext_vector_type(4))) int i32x4;

// ---------------------------------------------------------------------------
// Stage a 128x128 f32 panel (64 KB) from global into LDS.
// Preferred path: Tensor Data Mover (tensor_load_to_lds) issued by wave 0,
// completion via s_wait_tensorcnt + workgroup barrier.
// D# layout per CDNA5 ISA ch.8: group0 = {count/type, lds_addr, global_addr},
// group1 = {data_size, tensor_dim0/1, tile_dim0/1, tensor_dim0_stride}.
// This toolchain declares the 6-arg builtin:
//   (u32x4 g0, i32x8 g1, i32x4 g2, i32x4 g3, i32x8 g4, int cpol)
// ---------------------------------------------------------------------------
__device__ __forceinline__ void stage_B128(const float* __restrict__ B,
                                           float* __restrict__ Bs) {
#if __has_builtin(__builtin_amdgcn_tensor_load_to_lds) && __has_builtin(__builtin_amdgcn_s_wait_tensorcnt)
    if (threadIdx.x < 32) {
        const unsigned long long ga = (unsigned long long)(uintptr_t)B;
        const unsigned ldsoff = (unsigned)(uintptr_t)Bs;   // LDS aperture: low 32 bits = byte offset
        u32x4 g0;
        g0[0] = 1u;                                        // count=1, user descriptor
        g0[1] = ldsoff;                                    // lds_addr
        g0[2] = (unsigned)ga;                              // global_addr[31:0]
        g0[3] = (unsigned)((ga >> 32) & 0x01FFFFFFull)     // global_addr[56:32]
              | (2u << 30);                                // type=2 ("image")
        i32x8 g1;
        g1[0] = (int)(2u << 16);                           // data_size = 4 bytes
        g1[1] = (int)(128u << 16);                         // tensor_dim0[15:0] (=128)
        g1[2] = (int)(128u << 16);                         // tensor_dim0 hi=0 | tensor_dim1 lo (=128)
        g1[3] = (int)(128u << 16);                         // tensor_dim1 hi=0 | tile_dim0 (=128)
        g1[4] = (int)128u;                                 // tile_dim1 = 128, tile_dim2 = 0
        g1[5] = (int)128u;                                 // tensor_dim0_stride[31:0] = 128 elems
        g1[6] = 0;                                         // stride hi, dim1_stride lo
        g1[7] = 0;
        const i32x4 gz4 = {0, 0, 0, 0};                    // groups 2/3 unused (2D tensor)
        const i32x8 gz8 = {0, 0, 0, 0, 0, 0, 0, 0};
        __builtin_amdgcn_tensor_load_to_lds(g0, g1, gz4, gz4, gz8, 0);
        __builtin_amdgcn_s_wait_tensorcnt(0);
    }
    __syncthreads();
#else
    const int t = threadIdx.x;                             // 128 threads
    const float4* s4 = reinterpret_cast<const float4*>(B);
    float4* d4 = reinterpret_cast<float4*>(Bs);
#pragma unroll
    for (int i = 0; i < 32; ++i) d4[t + i * 128] = s4[t + i * 128];
    __syncthreads();
#endif
}

// ---------------------------------------------------------------------------
// Degree accumulation: one thread per edge, native f32 atomics.
// ---------------------------------------------------------------------------
__global__ void __launch_bounds__(256)
deg_kernel(const int* __restrict__ src, const int* __restrict__ dst,
           float* __restrict__ deg_out, float* __restrict__ deg_in, int E) {
    int e = blockIdx.x * blockDim.x + threadIdx.x;
    if (e < E) {
        unsafeAtomicAdd(&deg_out[src[e]], 1.0f);
        unsafeAtomicAdd(&deg_in[dst[e]], 1.0f);
    }
}

// d[i] = rsqrt(max(d[i], 1))
__global__ void __launch_bounds__(256)
rsqrt_kernel(float* __restrict__ d, int n) {
    int i = blockIdx.x * blockDim.x + threadIdx.x;
    if (i < n) d[i] = rsqrtf(fmaxf(d[i], 1.0f));
}

// ---------------------------------------------------------------------------
// C[m, 0:128] = (A[m, 0:128] @ B[128,128]) * rowscale[m]
// Block = 128 threads = 4 waves; block owns 16 rows; wave w owns cols
// 32w..32w+31 (two 16x16 tiles). V_WMMA_F32_16X16X4_F32, K stepped by 4.
// Branch-free inner loop: A row pointer clamped (out-of-range rows are
// computed but never stored), A fragment loaded as one b64.
// ---------------------------------------------------------------------------
__global__ void __launch_bounds__(128)
gemm128_scale_kernel(const float* __restrict__ A,        // M x 128
                     const float* __restrict__ B,        // 128 x 128
                     const float* __restrict__ rowscale, // M
                     float* __restrict__ C,              // M x 128
                     int M) {
    __shared__ float Bs[128 * 128];
    stage_B128(B, Bs);

    const int wave = threadIdx.x >> 5;
    const int lane = threadIdx.x & 31;
    const int half = lane >> 4;
    const int l16  = lane & 15;
    const int mbase = blockIdx.x * 16;
    const int cbase = wave * 32;

    int m = mbase + l16;
    if (m >= M) m = M - 1;                    // clamp: loads stay in-bounds
    const float* Arow = A + (size_t)m * 128;

    v8f acc0 = {}; v8f acc1 = {};
#pragma unroll
    for (int k = 0; k < 128; k += 4) {
        const int ka = k + 2 * half;
        const v2f a = *reinterpret_cast<const v2f*>(Arow + ka);
        v2f b0, b1;
        b0.x = Bs[ka * 128 + cbase + l16];
        b0.y = Bs[(ka + 1) * 128 + cbase + l16];
        b1.x = Bs[ka * 128 + cbase + 16 + l16];
        b1.y = Bs[(ka + 1) * 128 + cbase + 16 + l16];
        acc0 = __builtin_amdgcn_wmma_f32_16x16x4_f32(false, a, false, b0,
                                                     (short)0, acc0, false, false);
        acc1 = __builtin_amdgcn_wmma_f32_16x16x4_f32(false, a, false, b1,
                                                     (short)0, acc1, false, false);
    }
#pragma unroll
    for (int i = 0; i < 8; ++i) {
        const int mr = mbase + i + 8 * half;
        if (mr < M) {
            const float s = rowscale[mr];
            C[(size_t)mr * 128 + cbase + l16]      = acc0[i] * s;
            C[(size_t)mr * 128 + cbase + 16 + l16] = acc1[i] * s;
        }
    }
}

// ---------------------------------------------------------------------------
// Edge scatter: one wave per edge; lane gathers float4 of h[src], atomically
// accumulates into z_r[dst]. 4 edges per wave to amortize launch.
// ---------------------------------------------------------------------------
#define EPW 4
__global__ void __launch_bounds__(256)
scatter_kernel(const float* __restrict__ h,   // N x 128
               const int* __restrict__ src, const int* __restrict__ dst,
               float* __restrict__ zr,        // N x 128, accumulated
               int E) {
    const int gw   = (blockIdx.x * blockDim.x + threadIdx.x) >> 5;
    const int lane = threadIdx.x & 31;
    const int e0 = gw * EPW;
#pragma unroll
    for (int i = 0; i < EPW; ++i) {
        const int e = e0 + i;
        if (e >= E) break;
        const int s = src[e];
        const int d = dst[e];
        const float4 v = reinterpret_cast<const float4*>(h + (size_t)s * 128)[lane];
        float* dp = zr + (size_t)d * 128 + lane * 4;
        unsafeAtomicAdd(dp + 0, v.x);
        unsafeAtomicAdd(dp + 1, v.y);
        unsafeAtomicAdd(dp + 2, v.z);
        unsafeAtomicAdd(dp + 3, v.w);
    }
}

// z[(r*N+n)*128 + d] *= rs_in[r*N+n]   (float4 vectorized)
__global__ void __launch_bounds__(256)
scale_z_kernel(float* __restrict__ z, const float* __restrict__ rs_in, int N) {
    const long long i = (long long)blockIdx.x * blockDim.x + threadIdx.x;
    const long long total = 3LL * N * 32;   // float4 count
    if (i < total) {
        const float s = rs_in[i >> 5];
        float4* p = reinterpret_cast<float4*>(z) + i;
        float4 v = *p;
        v.x *= s; v.y *= s; v.z *= s; v.w *= s;
        *p = v;
    }
}

// ---------------------------------------------------------------------------
// Semantic attention logits:
//   proj = tanh(Z @ aW + ab); wsum[r] += sum over rows of relation r of proj.aq
// Z is (3*N) x 128 with row = r*N + n. Same WMMA tile structure; epilogue does
// tanh + aq scale, 16-lane shfl reduction, LDS combine, one atomic per row.
// ---------------------------------------------------------------------------
__global__ void __launch_bounds__(128)
attn_kernel(const float* __restrict__ Z,   // (3N) x 128
            const float* __restrict__ aW,  // 128 x 128
            const float* __restrict__ ab,  // 128
            const float* __restrict__ aq,  // 128
            float* __restrict__ wsum,      // 3
            int N) {
    __shared__ float Bs[128 * 128];
    __shared__ float rowsum[16];
    stage_B128(aW, Bs);

    const int M2 = 3 * N;
    const int wave = threadIdx.x >> 5;
    const int lane = threadIdx.x & 31;
    const int half = lane >> 4;
    const int l16  = lane & 15;
    const int mbase = blockIdx.x * 16;
    const int cbase = wave * 32;

    int m = mbase + l16;
    if (m >= M2) m = M2 - 1;                  // clamp
    const float* Arow = Z + (size_t)m * 128;

    v8f acc0 = {}; v8f acc1 = {};
#pragma unroll
    for (int k = 0; k < 128; k += 4) {
        const int ka = k + 2 * half;
        const v2f a = *reinterpret_cast<const v2f*>(Arow + ka);
        v2f b0, b1;
        b0.x = Bs[ka * 128 + cbase + l16];
        b0.y = Bs[(ka + 1) * 128 + cbase + l16];
        b1.x = Bs[ka * 128 + cbase + 16 + l16];
        b1.y = Bs[(ka + 1) * 128 + cbase + 16 + l16];
        acc0 = __builtin_amdgcn_wmma_f32_16x16x4_f32(false, a, false, b0,
                                                     (short)0, acc0, false, false);
        acc1 = __builtin_amdgcn_wmma_f32_16x16x4_f32(false, a, false, b1,
                                                     (short)0, acc1, false, false);
    }

    if (threadIdx.x < 16) rowsum[threadIdx.x] = 0.0f;
    __syncthreads();

    const int n0 = cbase + l16;
    const int n1 = cbase + 16 + l16;
    const float ab0 = ab[n0], ab1 = ab[n1];
    const float aq0 = aq[n0], aq1 = aq[n1];
#pragma unroll
    for (int i = 0; i < 8; ++i) {
        float v = tanhf(acc0[i] + ab0) * aq0 + tanhf(acc1[i] + ab1) * aq1;
#pragma unroll
        for (int off = 1; off < 16; off <<= 1)
            v += __shfl_xor(v, off, 32);      // stays within 16-lane group
        if (l16 == 0) atomicAdd(&rowsum[i + 8 * half], v);
    }
    __syncthreads();

    if (threadIdx.x < 16) {
        const int mr = mbase + threadIdx.x;
        if (mr < M2) unsafeAtomicAdd(&wsum[mr / N], rowsum[threadIdx.x]);
    }
}

// out[n,:] = sum_r softmax(wsum/N)_r * z[r,n,:]   (float4 vectorized)
__global__ void __launch_bounds__(256)
out_kernel(const float* __restrict__ z, const float* __restrict__ wsum,
           float* __restrict__ out, int N) {
    const long long i = (long long)blockIdx.x * blockDim.x + threadIdx.x;
    const long long total = (long long)N * 32;   // float4 count per relation
    if (i >= total) return;
    const float invN = 1.0f / (float)N;
    const float w0 = wsum[0] * invN, w1 = wsum[1] * invN, w2 = wsum[2] * invN;
    const float mx = fmaxf(w0, fmaxf(w1, w2));
    const float e0 = __expf(w0 - mx), e1 = __expf(w1 - mx), e2 = __expf(w2 - mx);
    const float inv = 1.0f / (e0 + e1 + e2);
    const float b0 = e0 * inv, b1 = e1 * inv, b2 = e2 * inv;
    const float4* zp = reinterpret_cast<const float4*>(z);
    const float4 v0 = zp[i], v1 = zp[i + total], v2 = zp[i + 2 * total];
    float4 o;
    o.x = b0 * v0.x + b1 * v1.x + b2 * v2.x;
    o.y = b0 * v0.y + b1 * v1.y + b2 * v2.y;
    o.z = b0 * v0.z + b1 * v1.z + b2 * v2.z;
    o.w = b0 * v0.w + b1 * v1.w + b2 * v2.w;
    reinterpret_cast<float4*>(out)[i] = o;
}

// ---------------------------------------------------------------------------
// Host-side orchestration.
// Input order (setup_inputs dict): x, src0,dst0,W0, src1,dst1,W1,
//                                  src2,dst2,W2, aW, ab, aq
// ---------------------------------------------------------------------------
extern "C" void kernel_launch(void* const* d_in, const int* in_sizes, int n_in,
                              void* d_out, int out_size, void* d_ws, size_t ws_size,
                              hipStream_t stream) {
    const float* x  = (const float*)d_in[0];
    const int*   src[3] = {(const int*)d_in[1], (const int*)d_in[4], (const int*)d_in[7]};
    const int*   dst[3] = {(const int*)d_in[2], (const int*)d_in[5], (const int*)d_in[8]};
    const float* W[3]   = {(const float*)d_in[3], (const float*)d_in[6], (const float*)d_in[9]};
    const float* aW = (const float*)d_in[10];
    const float* ab = (const float*)d_in[11];
    const float* aq = (const float*)d_in[12];
    float* out = (float*)d_out;

    const int N = in_sizes[0] / 128;   // feature dim D = 128
    const int E = in_sizes[1];

    // Workspace layout (floats): deg[6N] | wsum[16 pad] | z[3*N*128] | h[N*128]
    float* ws   = (float*)d_ws;
    float* deg  = ws;                  // deg_out r: deg + r*N ; deg_in r: deg + 3N + r*N
    float* wsum = ws + (size_t)6 * N;
    float* z    = wsum + 16;
    float* h    = z + (size_t)3 * N * 128;

    // Zero deg + wsum + z (accumulated into).
    const size_t zero_bytes = ((size_t)6 * N + 16 + (size_t)3 * N * 128) * sizeof(float);
    (void)hipMemsetAsync(d_ws, 0, zero_bytes, stream);

    // Degrees -> rsqrt(max(deg,1)).
    const int dblocks = (E + 255) / 256;
    for (int r = 0; r < 3; ++r)
        deg_kernel<<<dblocks, 256, 0, stream>>>(src[r], dst[r],
                                                deg + (size_t)r * N,
                                                deg + (size_t)(3 + r) * N, E);
    rsqrt_kernel<<<(6 * N + 255) / 256, 256, 0, stream>>>(deg, 6 * N);

    // Per-relation: h = (x @ W_r) * rsqrt(deg_out); scatter h[src] += into z_r.
    const int gblocks = (N + 15) / 16;
    const int sblocks = (E + 8 * EPW - 1) / (8 * EPW);   // 8 waves/block, EPW edges/wave
    for (int r = 0; r < 3; ++r) {
        gemm128_scale_kernel<<<gblocks, 128, 0, stream>>>(
            x, W[r], deg + (size_t)r * N, h, N);
        scatter_kernel<<<sblocks, 256, 0, stream>>>(
            h, src[r], dst[r], z + (size_t)r * N * 128, E);
    }

    // z *= rsqrt(deg_in) rowwise.
    {
        const long long total4 = 3LL * N * 32;
        scale_z_kernel<<<(int)((total4 + 255) / 256), 256, 0, stream>>>(
            z, deg + (size_t)3 * N, N);
    }

    // Semantic attention logits via WMMA, then softmax-weighted combine.
    attn_kernel<<<(3 * N + 15) / 16, 128, 0, stream>>>(z, aW, ab, aq, wsum, N);

    {
        const long long total4 = (long long)N * 32;
        out_kernel<<<(int)((total4 + 255) / 256), 256, 0, stream>>>(z, wsum, out, N);
    }
}